// VectorQuantizer_68324339745376
// MI455X (gfx1250) — compile-verified
//
#include <hip/hip_runtime.h>

typedef __attribute__((ext_vector_type(2))) float v2f;
typedef __attribute__((ext_vector_type(8))) float v8f;

#define K_EMB 512
#define D_EMB 64
#define N_ROWS 65536            // 64*32*32
#define OUT_ELEMS (N_ROWS * D_EMB)
#define LDS_STRIDE 68           // 64 + 4 pad: lane n hits bank (4n+c)%64 -> conflict-free ds_load_b64
#define WAVES_PER_BLOCK 4
#define GATHER_BLOCKS (OUT_ELEMS / 256)   // 16384

// ---------------- kernel 0: zero the histogram ----------------
__global__ void vq_init(int* __restrict__ counts) {
  const int t = threadIdx.x;
  if (t < K_EMB) counts[t] = 0;
}

// ---------------- kernel 1: fp32 WMMA nearest-codebook search ----------------
// score[m][k] = 2*x_m . e_k - |e_k|^2  (argmax == argmin distance)
__global__ __launch_bounds__(WAVES_PER_BLOCK * 32) void vq_argmin(
    const float* __restrict__ x, const float* __restrict__ emb, int* __restrict__ idx) {
  extern __shared__ float lds[];
  float* elds = lds;                         // [512][LDS_STRIDE]
  float* nlds = lds + K_EMB * LDS_STRIDE;    // [512] |e|^2

  const int t = threadIdx.x;

  // Stage full codebook (128 KB) into LDS, bank-padded rows, float4 granularity.
  for (int i = t; i < K_EMB * (D_EMB / 4); i += WAVES_PER_BLOCK * 32) {
    const int row = i >> 4;
    const int c = (i & 15) << 2;
    const float4 v = ((const float4*)emb)[i];
    *(float4*)(&elds[row * LDS_STRIDE + c]) = v;
  }
  __syncthreads();
  // Per-codeword squared norms.
  for (int r = t; r < K_EMB; r += WAVES_PER_BLOCK * 32) {
    float s = 0.f;
#pragma unroll
    for (int d = 0; d < D_EMB; ++d) {
      const float e = elds[r * LDS_STRIDE + d];
      s += e * e;
    }
    nlds[r] = s;
  }
  __syncthreads();

  const int wave = t >> 5;
  const int lane = t & 31;
  const int n = lane & 15;   // A: row-in-tile; B/D: column-in-tile
  const int h = lane >> 4;   // wave half
  const int rowBase = (blockIdx.x * WAVES_PER_BLOCK + wave) * 16;

  // A operand (16x4 fp32 per chunk): lane holds A[n][4*kc + 2*h + {0,1}], pre-scaled by 2.
  v2f a[16];
  const float* xr = x + (size_t)(rowBase + n) * D_EMB + 2 * h;
#pragma unroll
  for (int kc = 0; kc < 16; ++kc) {
    const v2f v = *(const v2f*)(xr + kc * 4);
    a[kc] = v + v;
  }

  float best[8];
  int bidx[8];
#pragma unroll
  for (int r = 0; r < 8; ++r) { best[r] = -3.402823e38f; bidx[r] = 0; }

  // Software-pipelined B operand: preload the next n-tile pair while the current
  // 32-WMMA chain executes, so ds_load latency is covered by matrix work.
  v2f cur0[16], cur1[16];
  {
    const float* b0 = &elds[(0 * 16 + n) * LDS_STRIDE + 2 * h];
    const float* b1 = &elds[(1 * 16 + n) * LDS_STRIDE + 2 * h];
#pragma unroll
    for (int kc = 0; kc < 16; ++kc) {
      cur0[kc] = *(const v2f*)(b0 + kc * 4);
      cur1[kc] = *(const v2f*)(b1 + kc * 4);
    }
  }

  for (int nt = 0; nt < K_EMB / 16; nt += 2) {
    const int n0 = nt * 16 + n;
    const int n1 = n0 + 16;
    const float nn0 = -nlds[n0];
    const float nn1 = -nlds[n1];
    v8f acc0 = {nn0, nn0, nn0, nn0, nn0, nn0, nn0, nn0};
    v8f acc1 = {nn1, nn1, nn1, nn1, nn1, nn1, nn1, nn1};

    // Issue next-iteration B loads first (wrap on last iter: harmless redundant load,
    // keeps control flow uniform and branch-free).
    const int ntn = (nt + 2 < K_EMB / 16) ? (nt + 2) : 0;
    v2f nxt0[16], nxt1[16];
    {
      const float* b0n = &elds[(ntn * 16 + n) * LDS_STRIDE + 2 * h];
      const float* b1n = &elds[((ntn + 1) * 16 + n) * LDS_STRIDE + 2 * h];
#pragma unroll
      for (int kc = 0; kc < 16; ++kc) {
        nxt0[kc] = *(const v2f*)(b0n + kc * 4);
        nxt1[kc] = *(const v2f*)(b1n + kc * 4);
      }
    }

    // Two independent accumulator chains hide WMMA->WMMA RAW latency.
#pragma unroll
    for (int kc = 0; kc < 16; ++kc) {
      acc0 = __builtin_amdgcn_wmma_f32_16x16x4_f32(false, a[kc], false, cur0[kc],
                                                   (short)0, acc0, false, false);
      acc1 = __builtin_amdgcn_wmma_f32_16x16x4_f32(false, a[kc], false, cur1[kc],
                                                   (short)0, acc1, false, false);
    }

    // D layout: VGPR r = row (r + 8h), lane%16 = column. Strict '>' keeps lowest index (argmin-first).
#pragma unroll
    for (int r = 0; r < 8; ++r) {
      if (acc0[r] > best[r]) { best[r] = acc0[r]; bidx[r] = n0; }
      if (acc1[r] > best[r]) { best[r] = acc1[r]; bidx[r] = n1; }
    }

#pragma unroll
    for (int kc = 0; kc < 16; ++kc) {
      cur0[kc] = nxt0[kc];
      cur1[kc] = nxt1[kc];
    }
  }

  // Cross-lane argmax within each 16-lane half (rows r+8h live in half h).
#pragma unroll
  for (int r = 0; r < 8; ++r) {
    float bs = best[r];
    int bi = bidx[r];
#pragma unroll
    for (int off = 8; off >= 1; off >>= 1) {
      const float os = __shfl_xor(bs, off, 16);
      const int oi = __shfl_xor(bi, off, 16);
      if (os > bs || (os == bs && oi < bi)) { bs = os; bi = oi; }
    }
    best[r] = bs;
    bidx[r] = bi;
  }
  if (n == 0) {
#pragma unroll
    for (int r = 0; r < 8; ++r) idx[rowBase + 8 * h + r] = bidx[r];
  }
}

// ---------------- kernel 2: gather + STE output + SSE partials + histogram ----------------
__global__ __launch_bounds__(256) void vq_gather(
    const float* __restrict__ x, const float* __restrict__ emb,
    const int* __restrict__ idx, float* __restrict__ out,
    int* __restrict__ counts, float* __restrict__ partials) {
  __shared__ float red[8];
  const int gid = blockIdx.x * 256 + threadIdx.x;
  const int row = gid >> 6;
  const int d = gid & 63;
  const int k = idx[row];
  const float q = emb[k * D_EMB + d];
  const float diff = q - x[gid];
  out[gid] = q;  // x + stop_grad(q - x) == q in forward value
  if (d == 0) atomicAdd(&counts[k], 1);

  float v = diff * diff;
#pragma unroll
  for (int off = 16; off >= 1; off >>= 1) v += __shfl_xor(v, off, 32);
  if ((threadIdx.x & 31) == 0) red[threadIdx.x >> 5] = v;
  __syncthreads();
  if (threadIdx.x == 0) {
    float s = 0.f;
#pragma unroll
    for (int w = 0; w < 8; ++w) s += red[w];
    partials[blockIdx.x] = s;  // deterministic: no float atomics
  }
}

// ---------------- kernel 3: loss + perplexity ----------------
__global__ __launch_bounds__(512) void vq_finalize(
    const int* __restrict__ counts, const float* __restrict__ partials,
    float* __restrict__ out) {
  __shared__ float redH[16];
  __shared__ float redS[16];
  const int t = threadIdx.x;

  const float p = (float)counts[t] * (1.0f / (float)N_ROWS);
  float H = p * logf(p + 1e-10f);
  float S = 0.f;
#pragma unroll
  for (int j = 0; j < GATHER_BLOCKS / 512; ++j) S += partials[t + 512 * j];

#pragma unroll
  for (int off = 16; off >= 1; off >>= 1) {
    H += __shfl_xor(H, off, 32);
    S += __shfl_xor(S, off, 32);
  }
  if ((t & 31) == 0) { redH[t >> 5] = H; redS[t >> 5] = S; }
  __syncthreads();
  if (t == 0) {
    float Hs = 0.f, Ss = 0.f;
#pragma unroll
    for (int w = 0; w < 16; ++w) { Hs += redH[w]; Ss += redS[w]; }
    // loss = emb_loss + 0.25*commit_loss = 1.25 * mean((q-x)^2)
    out[OUT_ELEMS] = 1.25f * Ss * (1.0f / (float)OUT_ELEMS);
    out[OUT_ELEMS + 1] = expf(-Hs);
  }
}

extern "C" void kernel_launch(void* const* d_in, const int* in_sizes, int n_in,
                              void* d_out, int out_size, void* d_ws, size_t ws_size,
                              hipStream_t stream) {
  (void)in_sizes; (void)n_in; (void)out_size; (void)ws_size;
  const float* x = (const float*)d_in[0];    // [65536, 64]
  const float* emb = (const float*)d_in[1];  // [512, 64]
  float* out = (float*)d_out;                // [65536*64] q  + [1] loss + [1] perplexity

  char* ws = (char*)d_ws;
  int* counts = (int*)ws;                                   // 512 ints
  float* partials = (float*)(ws + 2048);                    // 16384 floats
  int* idx = (int*)(ws + 2048 + GATHER_BLOCKS * 4);         // 65536 ints

  vq_init<<<1, 512, 0, stream>>>(counts);

  const size_t ldsBytes = (size_t)(K_EMB * LDS_STRIDE + K_EMB) * sizeof(float); // 141312 B
  vq_argmin<<<N_ROWS / (16 * WAVES_PER_BLOCK), WAVES_PER_BLOCK * 32, ldsBytes, stream>>>(
      x, emb, idx);

  vq_gather<<<GATHER_BLOCKS, 256, 0, stream>>>(x, emb, idx, out, counts, partials);

  vq_finalize<<<1, 512, 0, stream>>>(counts, partials, out);
}